// _BaseNet_335007449683
// MI455X (gfx1250) — compile-verified
//
#include <hip/hip_runtime.h>
#include <cstdint>
#include <cstddef>

// Problem constants (from the reference)
#define BB   16
#define CC   80
#define HH   128
#define WW   128
#define KK   100
#define HWQ  (HH * WW)            // 16384
#define SCL  4.0f                 // 512/128

// NMS tile geometry: 256 threads -> 32x8 outputs.
// LDS tile: 10 rows x 40 cols; aligned 32-float span at cols [4,36),
// left halo at col 3, right halo at col 36. Row stride 160B (16B aligned).
#define TW   32
#define TH   8
#define LROW 40
#define LNR  (TH + 2)             // 10 rows

#define NB1  512                  // level-1 buckets (bits >> 21; values < 1.0 -> < 508)
#define NB2  2048                 // level-2 buckets (bits [20:10])
#define CAP  4096                 // per-batch candidate cap

// Workspace layout (ints):
//   [0..15] counters  [16..31] t1  [32..47] above1  [48..63] thresh(bits)
//   [64..]  hist1 16*512 ; then hist2 16*2048 ; then uint2 cands 16*4096
#define WS_ZERO_INTS (64 + BB * NB1 + BB * NB2)   // 41024

__global__ __launch_bounds__(256) void zero_ws_kernel(int* ws, int n) {
    int i = blockIdx.x * 256 + threadIdx.x;
    if (i < n) ws[i] = 0;
}

__device__ __forceinline__ int clampi(int v, int lo, int hi) {
    return v < lo ? lo : (v > hi ? hi : v);
}

// Fused 3x3 local-max NMS over one (b,c) 128x128 plane tile, staged through
// LDS with CDNA5 async global->LDS copies (ASYNCcnt). Replicate-clamp halo is
// exactly equivalent to the reference's -inf-padded reduce_window max.
template <int PHASE>
__global__ __launch_bounds__(256) void nms_pass_kernel(const float* __restrict__ heat,
                                                       int* __restrict__ hist1,
                                                       int* __restrict__ hist2,
                                                       const int* __restrict__ t1,
                                                       const unsigned* __restrict__ thresh,
                                                       int* __restrict__ counters,
                                                       uint2* __restrict__ cands) {
    __shared__ __align__(16) float tile[LNR * LROW];
    __shared__ int lhist[NB1];

    const int tid = threadIdx.x;
    const int tx0 = (blockIdx.x % (WW / TW)) * TW;
    const int ty0 = (blockIdx.x / (WW / TW)) * TH;
    const int c   = blockIdx.y;
    const int b   = blockIdx.z;
    const float* plane = heat + (size_t)(b * CC + c) * HWQ;

    if (PHASE == 1) {
        lhist[tid] = 0;
        lhist[tid + 256] = 0;
    }

    // ---- async global -> LDS tile stage: 80x B128 (interior) + 20x B32 (x-halo)
    {
        const int e = tid;
        if (e < 80) {                      // row = e/8, 16B-aligned 4-float segment
            int row = e >> 3, seg = e & 7;
            int gy  = clampi(ty0 + row - 1, 0, HH - 1);
            unsigned long long gaddr =
                (unsigned long long)(uintptr_t)(plane + gy * WW + tx0 + seg * 4);
            // low 32 bits of generic shared pointer == LDS byte offset
            unsigned lds_addr = (unsigned)(uintptr_t)(&tile[row * LROW + 4 + seg * 4]);
            asm volatile("global_load_async_to_lds_b128 %0, %1, off"
                         :: "v"(lds_addr), "v"(gaddr) : "memory");
        } else if (e < 100) {              // left/right halo columns
            int h = e - 80, row = h >> 1, side = h & 1;
            int gy = clampi(ty0 + row - 1, 0, HH - 1);
            int gx = clampi(side ? (tx0 + TW) : (tx0 - 1), 0, WW - 1);
            unsigned long long gaddr =
                (unsigned long long)(uintptr_t)(plane + gy * WW + gx);
            unsigned lds_addr =
                (unsigned)(uintptr_t)(&tile[row * LROW + (side ? (TW + 4) : 3)]);
            asm volatile("global_load_async_to_lds_b32 %0, %1, off"
                         :: "v"(lds_addr), "v"(gaddr) : "memory");
        }
    }
    asm volatile("s_wait_asynccnt 0" ::: "memory");
    __syncthreads();

    // ---- 3x3 local max from LDS (center at column lx+4) ----
    const int lx = tid % TW;
    const int ly = tid / TW;
    const float* r0 = &tile[(ly + 0) * LROW + lx + 3];
    const float* r1 = &tile[(ly + 1) * LROW + lx + 3];
    const float* r2 = &tile[(ly + 2) * LROW + lx + 3];
    float ctr = r1[1];
    float m = fmaxf(fmaxf(fmaxf(r0[0], r0[1]), fmaxf(r0[2], r1[0])),
                    fmaxf(fmaxf(r1[1], r1[2]), fmaxf(fmaxf(r2[0], r2[1]), r2[2])));
    float nms = (m == ctr) ? ctr : 0.0f;
    unsigned bits = __float_as_uint(nms);

    if (PHASE == 1) {
        atomicAdd(&lhist[bits >> 21], 1);
        __syncthreads();
        for (int i = tid; i < NB1; i += 256)
            if (lhist[i]) atomicAdd(&hist1[b * NB1 + i], lhist[i]);
    } else if (PHASE == 2) {
        if ((int)(bits >> 21) == t1[b])
            atomicAdd(&hist2[b * NB2 + ((bits >> 10) & (NB2 - 1))], 1);
    } else {
        if (bits != 0u && bits >= thresh[b]) {
            int pos = atomicAdd(&counters[b], 1);
            if (pos < CAP) {
                unsigned idx = (unsigned)(c * HWQ + (ty0 + ly) * WW + (tx0 + lx));
                cands[b * CAP + pos] = make_uint2(bits, idx);
            }
        }
    }
}

__global__ void scan1_kernel(const int* __restrict__ hist1, int* t1, int* above1) {
    int b = threadIdx.x;
    if (b >= BB) return;
    int cum = 0, tb = 0, ab = 0;
    for (int i = NB1 - 1; i >= 0; --i) {
        int h = hist1[b * NB1 + i];
        if (cum + h >= KK) { tb = i; ab = cum; break; }
        cum += h;
    }
    t1[b] = tb;
    above1[b] = ab;
}

__global__ void scan2_kernel(const int* __restrict__ hist2, const int* __restrict__ t1,
                             const int* __restrict__ above1, unsigned* thresh) {
    int b = threadIdx.x;
    if (b >= BB) return;
    int need = KK - above1[b];
    int cum = 0, tb = 0;
    for (int i = NB2 - 1; i >= 0; --i) {
        int h = hist2[b * NB2 + i];
        if (cum + h >= need) { tb = i; break; }
        cum += h;
    }
    thresh[b] = (((unsigned)t1[b]) << 21) | (((unsigned)tb) << 10);
}

// Exact top-100 over <=4096 candidates per batch, then gather + bbox math.
__global__ __launch_bounds__(256) void finalize_kernel(const float* __restrict__ wh,
                                                       const float* __restrict__ off,
                                                       const uint2* __restrict__ cands,
                                                       const int* __restrict__ counters,
                                                       float* __restrict__ out_bbox,
                                                       int* __restrict__ out_lab) {
    __shared__ unsigned long long keys[CAP];
    __shared__ unsigned long long red[256];
    __shared__ unsigned sel_idx[KK];
    __shared__ unsigned sel_bits[KK];

    const int b = blockIdx.x;
    const int tid = threadIdx.x;
    int n = counters[b];
    if (n > CAP) n = CAP;

    for (int i = tid; i < CAP; i += 256) {
        unsigned long long k = 0ull;
        if (i < n) {
            uint2 cd = cands[b * CAP + i];
            // descending score, ascending-index tie-break (matches lax.top_k)
            k = ((unsigned long long)cd.x << 32) | (unsigned)(~cd.y);
        }
        keys[i] = k;
    }
    __syncthreads();

    for (int k = 0; k < KK; ++k) {
        unsigned long long bk = 0ull;
        for (int i = tid; i < CAP; i += 256) {
            unsigned long long v = keys[i];
            if (v > bk) bk = v;
        }
        red[tid] = bk;
        __syncthreads();
        for (int s = 128; s > 0; s >>= 1) {
            if (tid < s) {
                unsigned long long o = red[tid + s];
                if (o > red[tid]) red[tid] = o;
            }
            __syncthreads();
        }
        unsigned long long wk = red[0];
        if (tid == 0) {
            sel_bits[k] = (unsigned)(wk >> 32);
            sel_idx[k]  = ~(unsigned)wk;
        }
        if (wk != 0ull) {
            for (int i = tid; i < CAP; i += 256)
                if (keys[i] == wk) keys[i] = 0ull;
        }
        __syncthreads();
    }

    if (tid < KK) {
        unsigned bits = sel_bits[tid];
        unsigned idx  = sel_idx[tid];
        if (bits == 0u) idx = 0u;            // degenerate-safety (can't occur here)
        float score = __uint_as_float(bits);
        int label = (int)(idx / HWQ);
        int r     = (int)(idx % HWQ);
        int y = r / WW, x = r % WW;

        const float* whp = wh  + (size_t)b * 2 * HWQ;
        const float* ofp = off + (size_t)b * 2 * HWQ;
        float w_ = whp[r];
        float h_ = whp[HWQ + r];
        float ox = ofp[r];
        float oy = ofp[HWQ + r];

        float tx = (float)x + ox;
        float ty = (float)y + oy;
        float* o = out_bbox + (size_t)(b * KK + tid) * 5;
        o[0] = (tx - w_ * 0.5f) * SCL;
        o[1] = (ty - h_ * 0.5f) * SCL;
        o[2] = (tx + w_ * 0.5f) * SCL;
        o[3] = (ty + h_ * 0.5f) * SCL;
        o[4] = score;
        out_lab[b * KK + tid] = label;
    }
}

extern "C" void kernel_launch(void* const* d_in, const int* in_sizes, int n_in,
                              void* d_out, int out_size, void* d_ws, size_t ws_size,
                              hipStream_t stream) {
    (void)in_sizes; (void)n_in; (void)out_size; (void)ws_size;
    const float* heat = (const float*)d_in[0];
    const float* wh   = (const float*)d_in[1];
    const float* off  = (const float*)d_in[2];

    int* counters   = (int*)d_ws;
    int* t1         = counters + BB;
    int* above1     = t1 + BB;
    unsigned* thr   = (unsigned*)(above1 + BB);
    int* hist1      = (int*)(thr + BB);
    int* hist2      = hist1 + BB * NB1;
    uint2* cands    = (uint2*)(hist2 + BB * NB2);

    float* out_bbox = (float*)d_out;
    int*   out_lab  = (int*)d_out + BB * KK * 5;   // labels bit-packed after 8000 floats

    zero_ws_kernel<<<(WS_ZERO_INTS + 255) / 256, 256, 0, stream>>>((int*)d_ws, WS_ZERO_INTS);

    dim3 gridN((WW / TW) * (HH / TH), CC, BB);     // 64 x 80 x 16
    nms_pass_kernel<1><<<gridN, 256, 0, stream>>>(heat, hist1, hist2, t1, thr, counters, cands);
    scan1_kernel<<<1, BB, 0, stream>>>(hist1, t1, above1);
    nms_pass_kernel<2><<<gridN, 256, 0, stream>>>(heat, hist1, hist2, t1, thr, counters, cands);
    scan2_kernel<<<1, BB, 0, stream>>>(hist2, t1, above1, thr);
    nms_pass_kernel<3><<<gridN, 256, 0, stream>>>(heat, hist1, hist2, t1, thr, counters, cands);
    finalize_kernel<<<BB, 256, 0, stream>>>(wh, off, cands, counters, out_bbox, out_lab);
}